// GraphTransformerLayer_46007689675007
// MI455X (gfx1250) — compile-verified
//
#include <hip/hip_runtime.h>

#define NN 50000
#define NE 800000
#define EMB 64
#define HEADS 4
#define HC 256        // HEADS*EMB
#define EDIM 16
#define NEG_SLOPE 0.2f
#define LN_EPS 1e-5f

typedef float v2f __attribute__((ext_vector_type(2)));
typedef float v8f __attribute__((ext_vector_type(8)));

// ---- ordered-uint encoding for float atomicMax ----
__device__ __forceinline__ unsigned enc_f32(float f) {
    unsigned u = __float_as_uint(f);
    return (u & 0x80000000u) ? ~u : (u | 0x80000000u);
}
__device__ __forceinline__ float dec_f32(unsigned u) {
    unsigned b = (u & 0x80000000u) ? (u & 0x7FFFFFFFu) : ~u;
    return __uint_as_float(b);
}

// ---------------------------------------------------------------------------
// 0) workspace zero-fill
// ---------------------------------------------------------------------------
__global__ void zero_fill_kernel(float* __restrict__ p, size_t n) {
    size_t i = (size_t)blockIdx.x * blockDim.x + threadIdx.x;
    if (i < n) p[i] = 0.0f;
}

// ---------------------------------------------------------------------------
// 1) xw = x @ W   [N,64] x [64,256] -> [N,256]   (V_WMMA_F32_16X16X4_F32)
//    one wave per 16x16 output tile; blockDim=(32,4): 4 col tiles per block
// ---------------------------------------------------------------------------
__global__ void gemm_xw_kernel(const float* __restrict__ x,
                               const float* __restrict__ W,
                               float* __restrict__ xw) {
    const int lane  = threadIdx.x;                          // 0..31
    const int ct    = blockIdx.y * blockDim.y + threadIdx.y; // col tile 0..15
    const int mt    = blockIdx.x;                            // row tile 0..3124
    const int l15   = lane & 15;
    const int khalf = lane >> 4;                             // 0 or 1
    const int row   = mt * 16 + l15;
    const int col   = ct * 16 + l15;
    v8f c = {};
#pragma unroll 4
    for (int k = 0; k < EMB; k += 4) {
        const int kb = k + khalf * 2;
        v2f a, b;
        const float2 av = *(const float2*)(x + (size_t)row * EMB + kb); // 8B aligned
        a[0] = av.x; a[1] = av.y;
        b[0] = W[(size_t)kb * HC + col];
        b[1] = W[(size_t)(kb + 1) * HC + col];
        c = __builtin_amdgcn_wmma_f32_16x16x4_f32(false, a, false, b,
                                                  (short)0, c, false, false);
    }
#pragma unroll
    for (int r = 0; r < 8; ++r) {
        const int mm = r + khalf * 8;
        xw[(size_t)(mt * 16 + mm) * HC + col] = c[r];
    }
}

// ---------------------------------------------------------------------------
// 2) per-node attention scalars: a_src[n,h], a_dst[n,h]
// ---------------------------------------------------------------------------
__global__ void node_att_kernel(const float* __restrict__ xw,
                                const float* __restrict__ att_src,
                                const float* __restrict__ att_dst,
                                float* __restrict__ a_src,
                                float* __restrict__ a_dst) {
    int t = blockIdx.x * blockDim.x + threadIdx.x;
    if (t >= NN * HEADS) return;
    const int n = t >> 2, h = t & 3;
    const float* xp = xw + (size_t)n * HC + h * EMB;
    const float* as = att_src + h * EMB;
    const float* ad = att_dst + h * EMB;
    float ss = 0.f, sd = 0.f;
#pragma unroll 8
    for (int cidx = 0; cidx < EMB; ++cidx) {
        const float v = xp[cidx];
        ss += v * as[cidx];
        sd += v * ad[cidx];
    }
    a_src[t] = ss;
    a_dst[t] = sd;
}

// ---------------------------------------------------------------------------
// 3) self-loop attr = mean of incoming edge_attr: scatter count + sum
// ---------------------------------------------------------------------------
__global__ void edge_meansum_kernel(const int* __restrict__ dst,
                                    const float* __restrict__ edge_attr,
                                    float* __restrict__ cnt,
                                    float* __restrict__ loop_sum) {
    int e = blockIdx.x * blockDim.x + threadIdx.x;
    if (e >= NE) return;
    const int d = dst[e];
    atomicAdd(&cnt[d], 1.0f);
    const float* ea = edge_attr + (size_t)e * EDIM;
    float* ls = loop_sum + (size_t)d * EDIM;
#pragma unroll
    for (int j = 0; j < EDIM; ++j) atomicAdd(&ls[j], ea[j]);
}

// ---------------------------------------------------------------------------
// 4) fold W_edge (·) att_edge into M[16,4]:  a_edge = ea @ M
//    (removes the [E+N,256] ew GEMM and ~870MB of HBM traffic)
// ---------------------------------------------------------------------------
__global__ void edge_mat_kernel(const float* __restrict__ W_edge,
                                const float* __restrict__ att_edge,
                                float* __restrict__ M) {
    const int t = threadIdx.x;        // 0..63
    const int d = t >> 2, h = t & 3;
    float s = 0.f;
#pragma unroll 8
    for (int cidx = 0; cidx < EMB; ++cidx)
        s += W_edge[(size_t)d * HC + h * EMB + cidx] * att_edge[h * EMB + cidx];
    M[d * HEADS + h] = s;
}

// ---------------------------------------------------------------------------
// 5) a_edge[e,h] for all E real edges + N self-loop edges
// ---------------------------------------------------------------------------
__global__ void a_edge_kernel(const float* __restrict__ edge_attr,
                              const float* __restrict__ loop_sum,
                              const float* __restrict__ cnt,
                              const float* __restrict__ M,
                              float* __restrict__ a_edge) {
    int e = blockIdx.x * blockDim.x + threadIdx.x;
    if (e >= NE + NN) return;
    float ea[EDIM];
    if (e < NE) {
#pragma unroll
        for (int j = 0; j < EDIM; ++j) ea[j] = edge_attr[(size_t)e * EDIM + j];
    } else {
        const int n = e - NE;
        const float inv = 1.0f / fmaxf(cnt[n], 1.0f);
#pragma unroll
        for (int j = 0; j < EDIM; ++j) ea[j] = loop_sum[(size_t)n * EDIM + j] * inv;
    }
#pragma unroll
    for (int h = 0; h < HEADS; ++h) {
        float s = 0.f;
#pragma unroll
        for (int j = 0; j < EDIM; ++j) s += ea[j] * M[j * HEADS + h];
        a_edge[(size_t)e * HEADS + h] = s;
    }
}

// ---------------------------------------------------------------------------
// 6) alpha = leaky_relu(a_src[s]+a_dst[d]+a_edge); segment max via atomicMax
// ---------------------------------------------------------------------------
__global__ void alpha_max_kernel(const int* __restrict__ src,
                                 const int* __restrict__ dst,
                                 const float* __restrict__ a_src,
                                 const float* __restrict__ a_dst,
                                 const float* __restrict__ a_edge,
                                 float* __restrict__ alpha,
                                 unsigned* __restrict__ amax) {
    int e = blockIdx.x * blockDim.x + threadIdx.x;
    if (e >= NE + NN) return;
    const int s = (e < NE) ? src[e] : (e - NE);
    const int d = (e < NE) ? dst[e] : (e - NE);
#pragma unroll
    for (int h = 0; h < HEADS; ++h) {
        float al = a_src[s * HEADS + h] + a_dst[d * HEADS + h] +
                   a_edge[(size_t)e * HEADS + h];
        al = (al >= 0.f) ? al : NEG_SLOPE * al;
        alpha[(size_t)e * HEADS + h] = al;
        atomicMax(&amax[d * HEADS + h], enc_f32(al));
    }
}

// ---------------------------------------------------------------------------
// 7) ae = exp(alpha - amax[d]); segment sum -> denom; alpha <- ae (in place)
// ---------------------------------------------------------------------------
__global__ void expsum_kernel(const int* __restrict__ dst,
                              const unsigned* __restrict__ amax,
                              float* __restrict__ alpha,
                              float* __restrict__ denom) {
    int e = blockIdx.x * blockDim.x + threadIdx.x;
    if (e >= NE + NN) return;
    const int d = (e < NE) ? dst[e] : (e - NE);
#pragma unroll
    for (int h = 0; h < HEADS; ++h) {
        const float ae = __expf(alpha[(size_t)e * HEADS + h] -
                                dec_f32(amax[d * HEADS + h]));
        alpha[(size_t)e * HEADS + h] = ae;
        atomicAdd(&denom[d * HEADS + h], ae);
    }
}

// ---------------------------------------------------------------------------
// 8) aggregation: one wave32 per edge; 8 floats/lane scatter-add into agg[dst]
//    (the bandwidth-dominant step: ~1.7GB -> ~75us at 23.3 TB/s)
// ---------------------------------------------------------------------------
__global__ void aggregate_kernel(const int* __restrict__ src,
                                 const int* __restrict__ dst,
                                 const float* __restrict__ xw,
                                 const float* __restrict__ alpha,
                                 const float* __restrict__ denom,
                                 float* __restrict__ agg) {
    const int lane = threadIdx.x & 31;
    const int wid  = threadIdx.x >> 5;
    const long e   = (long)blockIdx.x * (blockDim.x >> 5) + wid;
    if (e >= NE + NN) return;
    const int s = (e < NE) ? src[e] : (int)(e - NE);
    const int d = (e < NE) ? dst[e] : (int)(e - NE);
    float w[HEADS];
#pragma unroll
    for (int h = 0; h < HEADS; ++h)
        w[h] = alpha[(size_t)e * HEADS + h] / denom[d * HEADS + h];
    const float* xs = xw + (size_t)s * HC;
    float* ag = agg + (size_t)d * HC;
#pragma unroll
    for (int i = 0; i < 8; ++i) {
        const int idx = lane + i * 32;      // 0..255, head = idx>>6
        atomicAdd(&ag[idx], xs[idx] * w[idx >> 6]);
    }
}

// ---------------------------------------------------------------------------
// 9) proj = (agg + bias) @ Wp + bp   [N,256] x [256,64]  (f32 WMMA, K=256)
// ---------------------------------------------------------------------------
__global__ void gemm_proj_kernel(const float* __restrict__ agg,
                                 const float* __restrict__ bias,
                                 const float* __restrict__ Wp,
                                 const float* __restrict__ bp,
                                 float* __restrict__ proj) {
    const int lane  = threadIdx.x;       // 0..31
    const int ct    = threadIdx.y;       // 0..3 (64 cols = 4 tiles)
    const int mt    = blockIdx.x;        // 0..3124
    const int l15   = lane & 15;
    const int khalf = lane >> 4;
    const int row   = mt * 16 + l15;
    const int col   = ct * 16 + l15;
    v8f c = {};
    for (int k = 0; k < HC; k += 4) {
        const int kb = k + khalf * 2;
        v2f a, b;
        a[0] = agg[(size_t)row * HC + kb]     + bias[kb];
        a[1] = agg[(size_t)row * HC + kb + 1] + bias[kb + 1];
        b[0] = Wp[(size_t)kb * EMB + col];
        b[1] = Wp[(size_t)(kb + 1) * EMB + col];
        c = __builtin_amdgcn_wmma_f32_16x16x4_f32(false, a, false, b,
                                                  (short)0, c, false, false);
    }
#pragma unroll
    for (int r = 0; r < 8; ++r) {
        const int mm = r + khalf * 8;
        proj[(size_t)(mt * 16 + mm) * EMB + col] = c[r] + bp[col];
    }
}

// ---------------------------------------------------------------------------
// 10) h = x + elu(proj); LayerNorm(64) -> out.  One wave per row, 2 elems/lane.
// ---------------------------------------------------------------------------
__global__ void ln_kernel(const float* __restrict__ x,
                          const float* __restrict__ proj,
                          const float* __restrict__ gamma,
                          const float* __restrict__ beta,
                          float* __restrict__ out) {
    const int row = blockIdx.x * blockDim.y + threadIdx.y;
    if (row >= NN) return;
    const int lane = threadIdx.x;        // 0..31, handles cols lane, lane+32
    float p0 = proj[(size_t)row * EMB + lane];
    float p1 = proj[(size_t)row * EMB + lane + 32];
    p0 = (p0 > 0.f) ? p0 : (__expf(p0) - 1.0f);   // elu(alpha=1)
    p1 = (p1 > 0.f) ? p1 : (__expf(p1) - 1.0f);
    const float v0 = x[(size_t)row * EMB + lane] + p0;
    const float v1 = x[(size_t)row * EMB + lane + 32] + p1;
    float s = v0 + v1;
#pragma unroll
    for (int m = 16; m > 0; m >>= 1) s += __shfl_xor(s, m, 32);
    const float mu = s * (1.0f / 64.0f);
    const float d0 = v0 - mu, d1 = v1 - mu;
    float q = d0 * d0 + d1 * d1;
#pragma unroll
    for (int m = 16; m > 0; m >>= 1) q += __shfl_xor(q, m, 32);
    const float inv = rsqrtf(q * (1.0f / 64.0f) + LN_EPS);
    out[(size_t)row * EMB + lane]      = d0 * inv * gamma[lane]      + beta[lane];
    out[(size_t)row * EMB + lane + 32] = d1 * inv * gamma[lane + 32] + beta[lane + 32];
}

// ---------------------------------------------------------------------------
extern "C" void kernel_launch(void* const* d_in, const int* in_sizes, int n_in,
                              void* d_out, int out_size, void* d_ws, size_t ws_size,
                              hipStream_t stream) {
    const float* x         = (const float*)d_in[0];
    const int*   ei        = (const int*)  d_in[1];   // [2,E]
    const float* edge_attr = (const float*)d_in[2];
    const float* W         = (const float*)d_in[3];
    const float* W_edge    = (const float*)d_in[4];
    const float* att_src   = (const float*)d_in[5];
    const float* att_dst   = (const float*)d_in[6];
    const float* att_edge  = (const float*)d_in[7];
    const float* bias      = (const float*)d_in[8];
    const float* Wp        = (const float*)d_in[9];
    const float* bp        = (const float*)d_in[10];
    const float* gamma     = (const float*)d_in[11];
    const float* beta      = (const float*)d_in[12];
    const int* src  = ei;
    const int* dstp = ei + NE;

    // workspace layout (floats); zero-initialized region first (~150 MB total)
    float* ws = (float*)d_ws;
    size_t off = 0;
    float*    agg      = ws + off; off += (size_t)NN * HC;     // 12.8M
    float*    cnt      = ws + off; off += NN;                  // 50K
    float*    loop_sum = ws + off; off += (size_t)NN * EDIM;   // 800K
    unsigned* amax     = (unsigned*)(ws + off); off += (size_t)NN * HEADS; // 200K
    float*    denom    = ws + off; off += (size_t)NN * HEADS;  // 200K
    const size_t zeroN = off;
    float*    xw       = ws + off; off += (size_t)NN * HC;
    float*    a_src    = ws + off; off += (size_t)NN * HEADS;
    float*    a_dst    = ws + off; off += (size_t)NN * HEADS;
    float*    Mmat     = ws + off; off += EDIM * HEADS;
    float*    a_edge   = ws + off; off += (size_t)(NE + NN) * HEADS;
    float*    alpha    = ws + off; off += (size_t)(NE + NN) * HEADS;
    float*    proj     = ws + off; off += (size_t)NN * EMB;
    (void)ws_size; (void)in_sizes; (void)n_in; (void)out_size;

    const int ET = NE + NN;

    zero_fill_kernel<<<(unsigned)((zeroN + 511) / 512), 512, 0, stream>>>(ws, zeroN);

    // dense node transform via WMMA: 3125 row tiles x 16 col tiles
    gemm_xw_kernel<<<dim3(NN / 16, (HC / 16) / 4), dim3(32, 4), 0, stream>>>(x, W, xw);
    node_att_kernel<<<(NN * HEADS + 255) / 256, 256, 0, stream>>>(xw, att_src, att_dst,
                                                                  a_src, a_dst);
    edge_meansum_kernel<<<(NE + 255) / 256, 256, 0, stream>>>(dstp, edge_attr,
                                                              cnt, loop_sum);
    edge_mat_kernel<<<1, 64, 0, stream>>>(W_edge, att_edge, Mmat);
    a_edge_kernel<<<(ET + 255) / 256, 256, 0, stream>>>(edge_attr, loop_sum, cnt,
                                                        Mmat, a_edge);
    alpha_max_kernel<<<(ET + 255) / 256, 256, 0, stream>>>(src, dstp, a_src, a_dst,
                                                           a_edge, alpha, amax);
    expsum_kernel<<<(ET + 255) / 256, 256, 0, stream>>>(dstp, amax, alpha, denom);
    aggregate_kernel<<<(ET + 7) / 8, 256, 0, stream>>>(src, dstp, xw, alpha,
                                                       denom, agg);
    gemm_proj_kernel<<<dim3(NN / 16, 1), dim3(32, 4), 0, stream>>>(agg, bias, Wp,
                                                                   bp, proj);
    ln_kernel<<<(NN + 7) / 8, dim3(32, 8), 0, stream>>>(x, proj, gamma, beta,
                                                        (float*)d_out);
}